// NTM_36730560315413
// MI455X (gfx1250) — compile-verified
//
#include <hip/hip_runtime.h>
#include <math.h>

#define EPS 1e-8f

typedef float v2f __attribute__((ext_vector_type(2)));
typedef float v4f __attribute__((ext_vector_type(4)));
typedef float v8f __attribute__((ext_vector_type(8)));

__device__ __forceinline__ float sigmoidf_(float v) { return 1.0f / (1.0f + expf(-v)); }

// ---------------------------------------------------------------------------
// WMMA GEMV: y[r] = act( sum_h W[r*cols+h] * x[h] + bias[r] )
// One wave handles 16 rows via V_WMMA_F32_16X16X4_F32.
// A (16x4 f32): lane L<16 holds {W[r0+L][h], W[r0+L][h+1]}, lane L+16 holds
// {W[r0+L][h+2], W[r0+L][h+3]}  (ISA 7.12.2 32-bit A layout).
// B (4x16): x[h..h+3] replicated across all 16 columns -> every D column holds y.
// D: VGPR v, lanes 0-15 = row r0+v ; lanes 16-31 = row r0+8+v.
// ---------------------------------------------------------------------------
__global__ __launch_bounds__(256) void wmma_gemv_f32(
    const float* __restrict__ W, const float* __restrict__ bias,
    const float* __restrict__ x, float* __restrict__ y, int cols, int act) {
  const int wave  = (blockIdx.x * 256 + threadIdx.x) >> 5;
  const int lane  = threadIdx.x & 31;
  const int r0    = wave * 16;
  const int rsel  = lane & 15;   // which of the 16 rows this lane feeds into A
  const int khalf = lane >> 4;   // 0 -> K=0,1 ; 1 -> K=2,3
  const float* wrow = W + (size_t)(r0 + rsel) * (size_t)cols + 2 * khalf;
  const float* xk   = x + 2 * khalf;

  v8f c = {};
  for (int h = 0; h < cols; h += 4) {
    v2f a = *(const v2f*)(wrow + h);
    v2f b = *(const v2f*)(xk + h);
    c = __builtin_amdgcn_wmma_f32_16x16x4_f32(false, a, false, b,
                                              (short)0, c, false, false);
  }
  if (lane == 0 || lane == 16) {
    const int base = r0 + (lane >> 4) * 8;
#pragma unroll
    for (int v = 0; v < 8; ++v) {
      float val = c[v] + (bias ? bias[base + v] : 0.0f);
      if (act == 1) val = sigmoidf_(val);
      y[base + v] = val;
    }
  }
}

// ---------------------------------------------------------------------------
// Dominant kernel: e[r] = sigmoid( w_We[r,:] . co + w_be[r] ), r in [0, N*M).
// 1 GiB streaming read -> one wave per row, b128 non-temporal loads, lane-
// coalesced (lane l covers floats l*4 + 128*k). Bandwidth-bound; VALU FMA.
// ---------------------------------------------------------------------------
__global__ __launch_bounds__(256) void erase_gemv(
    const float* __restrict__ W, const float* __restrict__ b,
    const float* __restrict__ co, float* __restrict__ e) {
  const int row  = (blockIdx.x * 256 + threadIdx.x) >> 5;
  const int lane = threadIdx.x & 31;
  const float* wrow = W + (size_t)row * 1024u;
  float acc = 0.0f;
#pragma unroll
  for (int k = 0; k < 8; ++k) {
    const int off = lane * 4 + k * 128;
    v4f w  = __builtin_nontemporal_load((const v4f*)(wrow + off));
    v4f xv = *(const v4f*)(co + off);
    acc += w[0] * xv[0] + w[1] * xv[1] + w[2] * xv[2] + w[3] * xv[3];
  }
#pragma unroll
  for (int s = 16; s > 0; s >>= 1) acc += __shfl_xor(acc, s, 32);
  if (lane == 0) e[row] = sigmoidf_(acc + b[row]);
}

// ||memory[i,:]|| per row; one wave per row (512 floats -> 4x v4f per lane).
__global__ __launch_bounds__(256) void row_norms(
    const float* __restrict__ memory, float* __restrict__ norms) {
  const int row  = (blockIdx.x * 256 + threadIdx.x) >> 5;
  const int lane = threadIdx.x & 31;
  const float* m = memory + (size_t)row * 512u;
  float acc = 0.0f;
#pragma unroll
  for (int k = 0; k < 4; ++k) {
    v4f v = *(const v4f*)(m + lane * 4 + k * 128);
    acc += v[0] * v[0] + v[1] * v[1] + v[2] * v[2] + v[3] * v[3];
  }
#pragma unroll
  for (int s = 16; s > 0; s >>= 1) acc += __shfl_xor(acc, s, 32);
  if (lane == 0) norms[row] = sqrtf(acc);
}

// beta/g head scalars: block b in {0:rbeta,1:rg,2:wbeta,3:wg}; even -> softplus.
__global__ __launch_bounds__(256) void head_scalars(
    const float* __restrict__ co,
    const float* w0, const float* b0, const float* w1, const float* b1,
    const float* w2, const float* b2, const float* w3, const float* b3,
    float* __restrict__ scalars) {
  __shared__ float red[256];
  const float* w; const float* b;
  switch (blockIdx.x) {
    case 0: w = w0; b = b0; break;
    case 1: w = w1; b = b1; break;
    case 2: w = w2; b = b2; break;
    default: w = w3; b = b3; break;
  }
  const int t = threadIdx.x;
  float acc = 0.0f;
  for (int i = t; i < 1024; i += 256) acc += w[i] * co[i];
  red[t] = acc; __syncthreads();
  for (int s = 128; s > 0; s >>= 1) { if (t < s) red[t] += red[t + s]; __syncthreads(); }
  if (t == 0) {
    float v = red[0] + b[0];
    if ((blockIdx.x & 1) == 0) v = fmaxf(v, 0.0f) + log1pf(expf(-fabsf(v)));  // softplus
    else                       v = sigmoidf_(v);                               // sigmoid
    scalars[blockIdx.x] = v;
  }
}

// Cosine-similarity addressing + sharpened softmax + gate interpolation.
// One 512-thread block per head; both heads interpolate against prev_read
// (reference preserves that source bug).
__global__ __launch_bounds__(512) void address_kernel(
    const float* __restrict__ dot_r, const float* __restrict__ dot_w,
    const float* __restrict__ rk, const float* __restrict__ wk,
    const float* __restrict__ norms, const float* __restrict__ scalars,
    const float* __restrict__ prev,
    float* __restrict__ out_read, float* __restrict__ out_write) {
  __shared__ float red[512];
  const int t = threadIdx.x;
  const int head = blockIdx.x;                 // 0=read, 1=write
  const float* dot = head ? dot_w : dot_r;
  const float* k   = head ? wk : rk;
  const float beta = scalars[head * 2 + 0];
  const float g    = scalars[head * 2 + 1];
  float* out = head ? out_write : out_read;

  float kv = k[t];
  red[t] = kv * kv; __syncthreads();
  for (int s = 256; s > 0; s >>= 1) { if (t < s) red[t] += red[t + s]; __syncthreads(); }
  const float knorm = sqrtf(red[0]); __syncthreads();

  const float sim = dot[t] / (fmaxf(norms[t], EPS) * fmaxf(knorm, EPS));
  const float z = beta * sim;
  red[t] = z; __syncthreads();
  for (int s = 256; s > 0; s >>= 1) { if (t < s) red[t] = fmaxf(red[t], red[t + s]); __syncthreads(); }
  const float zmax = red[0]; __syncthreads();
  const float ez = expf(z - zmax);
  red[t] = ez; __syncthreads();
  for (int s = 256; s > 0; s >>= 1) { if (t < s) red[t] += red[t + s]; __syncthreads(); }
  const float wc = ez / red[0];
  out[t] = g * wc + (1.0f - g) * prev[t];
}

// out[j] = sum_i s[i] * Mat[i*512 + j]   (used for read_out and write_state@e)
__global__ __launch_bounds__(128) void col_weighted_sum(
    const float* __restrict__ Mat, const float* __restrict__ s,
    float* __restrict__ out) {
  const int j = blockIdx.x * 128 + threadIdx.x;
  float acc = 0.0f;
  for (int i = 0; i < 512; ++i) acc += s[i] * Mat[i * 512 + j];
  out[j] = acc;
}

__global__ __launch_bounds__(256) void concat_kernel(
    const float* __restrict__ co, const float* __restrict__ ro,
    float* __restrict__ fcin) {
  const int t = blockIdx.x * 256 + threadIdx.x;   // t < 1536
  fcin[t] = (t < 1024) ? co[t] : ro[t - 1024];
}

// new_memory[i,j] = memory[i,j]*(1 - we[j]) + ws[i]*a[j]
__global__ __launch_bounds__(256) void write_memory(
    const float* __restrict__ memory, const float* __restrict__ we,
    const float* __restrict__ ws_state, const float* __restrict__ a,
    float* __restrict__ nm) {
  const int idx = blockIdx.x * 256 + threadIdx.x;
  const int i = idx >> 9, j = idx & 511;
  nm[idx] = memory[idx] * (1.0f - we[j]) + ws_state[i] * a[j];
}

extern "C" void kernel_launch(void* const* d_in, const int* in_sizes, int n_in,
                              void* d_out, int out_size, void* d_ws, size_t ws_size,
                              hipStream_t stream) {
  (void)in_sizes; (void)n_in; (void)out_size; (void)ws_size;
  const float* x         = (const float*)d_in[0];
  const float* prev_read = (const float*)d_in[1];
  // d_in[2] prev_write_state: unused by reference outputs
  const float* memory    = (const float*)d_in[3];
  const float* Wc = (const float*)d_in[4];  const float* bc   = (const float*)d_in[5];
  const float* r_Wk = (const float*)d_in[6]; const float* r_bk = (const float*)d_in[7];
  const float* r_Wbeta = (const float*)d_in[8]; const float* r_bbeta = (const float*)d_in[9];
  const float* r_Wg = (const float*)d_in[10];  const float* r_bg = (const float*)d_in[11];
  // 12..15: r_Ws/r_bs/r_Wgamma/r_bgamma -> dead code, skipped (saves 8 MB reads)
  const float* w_Wk = (const float*)d_in[16]; const float* w_bk = (const float*)d_in[17];
  const float* w_Wbeta = (const float*)d_in[18]; const float* w_bbeta = (const float*)d_in[19];
  const float* w_Wg = (const float*)d_in[20];  const float* w_bg = (const float*)d_in[21];
  // 22..25: w_Ws/w_bs/w_Wgamma/w_bgamma -> dead code, skipped
  const float* w_We = (const float*)d_in[26]; const float* w_be = (const float*)d_in[27];
  const float* w_Wa = (const float*)d_in[28]; const float* w_ba = (const float*)d_in[29];
  const float* fc_W = (const float*)d_in[30]; const float* fc_b = (const float*)d_in[31];

  float* ws = (float*)d_ws;
  float* co       = ws + 0;      // 1024
  float* rk       = ws + 1024;   // 512
  float* wk       = ws + 1536;   // 512
  float* av       = ws + 2048;   // 512
  float* dot_r    = ws + 2560;   // 512
  float* dot_w    = ws + 3072;   // 512
  float* norms    = ws + 3584;   // 512
  float* scalars  = ws + 4096;   // 4: [rbeta, rg, wbeta, wg]
  float* read_out = ws + 4608;   // 512
  float* fcin     = ws + 5120;   // 1536
  float* wE       = ws + 6656;   // 512 (write_state @ e)

  float* out         = (float*)d_out;      // [1024]
  float* read_state  = out + 1024;         // [512]
  float* write_state = out + 1536;         // [512]
  float* new_mem     = out + 2048;         // [512*512]
  float* etmp        = out + 2048;         // stage e here, overwritten by new_mem

  // Independent prep
  row_norms<<<64, 256, 0, stream>>>(memory, norms);
  // Controller (WMMA)
  wmma_gemv_f32<<<8, 256, 0, stream>>>(Wc, bc, x, co, 1024, 0);
  // Dominant 1 GiB stream: e = sigmoid(w_We @ co + w_be)
  erase_gemv<<<32768, 256, 0, stream>>>(w_We, w_be, co, etmp);
  // Head projections (WMMA)
  wmma_gemv_f32<<<4, 256, 0, stream>>>(r_Wk, r_bk, co, rk, 1024, 0);
  wmma_gemv_f32<<<4, 256, 0, stream>>>(w_Wk, w_bk, co, wk, 1024, 0);
  wmma_gemv_f32<<<4, 256, 0, stream>>>(w_Wa, w_ba, co, av, 1024, 0);
  head_scalars<<<4, 256, 0, stream>>>(co, r_Wbeta, r_bbeta, r_Wg, r_bg,
                                      w_Wbeta, w_bbeta, w_Wg, w_bg, scalars);
  // memory @ k (WMMA, no bias)
  wmma_gemv_f32<<<4, 256, 0, stream>>>(memory, nullptr, rk, dot_r, 512, 0);
  wmma_gemv_f32<<<4, 256, 0, stream>>>(memory, nullptr, wk, dot_w, 512, 0);
  // Addressing (both heads; both interpolate vs prev_read_state per reference)
  address_kernel<<<2, 512, 0, stream>>>(dot_r, dot_w, rk, wk, norms, scalars,
                                        prev_read, read_state, write_state);
  // read_out = read_state @ memory
  col_weighted_sum<<<4, 128, 0, stream>>>(memory, read_state, read_out);
  // out = sigmoid(fc([co, read_out]))
  concat_kernel<<<6, 256, 0, stream>>>(co, read_out, fcin);
  wmma_gemv_f32<<<8, 256, 0, stream>>>(fc_W, fc_b, fcin, out, 1536, 1);
  // we = write_state @ e ; then memory rewrite (overwrites staged e)
  col_weighted_sum<<<4, 128, 0, stream>>>(etmp, write_state, wE);
  write_memory<<<1024, 256, 0, stream>>>(memory, wE, write_state, av, new_mem);
}